// AdaptiveMoodCoherenceHysteresis_69441031242203
// MI455X (gfx1250) — compile-verified
//
#include <hip/hip_runtime.h>
#include <math.h>

typedef float v2f __attribute__((ext_vector_type(2)));
typedef float v8f __attribute__((ext_vector_type(8)));

#define ALPHA_MIN_ 0.08f
#define ALPHA_MAX_ 0.45f
#define BETA_      0.12f
#define SIG_SLOPE_ 8.0f
#define SIG_OFF_   1.5f
#define EPS_       1e-12f

#define B_   8
#define F_   64
#define T_   2048
#define FG_  4            // groups of 16 channels
#define TS_  8            // T-slices
#define TSL_ (T_ / TS_)   // 256 samples per slice

// Kernel 1: each wave owns (batch b, 16-channel group fg, T-slice ts).
// V_WMMA_F32_16X16X4_F32 with B = ones acts as a fp32 accumulating reducer:
// D[m,n] = sum_k A[m,k] + C[m,n]. A-tile per iteration = 16 channels x 4
// consecutive t samples (lane L: row = L&15, K-pair = (L>>4)*2).
// Partial sums out: part[((b*FG+fg)*TS+ts)*16 + row]*2 + {0:cos,1:sin}.
__global__ __launch_bounds__(256)
void k1_plv_partial(const float* __restrict__ phases, float* __restrict__ part) {
  const int wave = (blockIdx.x * blockDim.x + threadIdx.x) >> 5;
  const int lane = threadIdx.x & 31;
  const int b  = wave >> 5;      // 0..7
  const int r  = wave & 31;
  const int fg = r >> 3;         // 0..3
  const int ts = r & 7;          // 0..7

  const int row = lane & 15;
  const int hh  = lane >> 4;     // K half: 0 -> K0,K1 ; 1 -> K2,K3
  const int f   = fg * 16 + row;
  const float* base = phases + (size_t)(b * F_ + f) * T_ + ts * TSL_ + hh * 2;

  const v2f ones = {1.0f, 1.0f};
  v8f c_cos = {};
  v8f c_sin = {};

  #pragma unroll 4
  for (int i = 0; i < TSL_ / 4; ++i) {
    const float2 p = *(const float2*)(base + i * 4);
    float s0, c0, s1, c1;
    __sincosf(p.x, &s0, &c0);
    __sincosf(p.y, &s1, &c1);
    v2f ac = {c0, c1};
    v2f as = {s0, s1};
    // 8-arg form: (neg_a, A, neg_b, B, c_mod, C, reuse_a, reuse_b)
    c_cos = __builtin_amdgcn_wmma_f32_16x16x4_f32(false, ac, false, ones,
                                                  (short)0, c_cos, false, false);
    c_sin = __builtin_amdgcn_wmma_f32_16x16x4_f32(false, as, false, ones,
                                                  (short)0, c_sin, false, false);
  }

  // C/D layout: VGPR j, lanes 0-15 -> row j; lanes 16-31 -> row j+8.
  // All 16 columns are identical (B = ones), so lane 0 / lane 16 suffice.
  float* outp = part + (size_t)((b * FG_ + fg) * TS_ + ts) * 16 * 2;
  if (lane == 0) {
    #pragma unroll
    for (int j = 0; j < 8; ++j) {
      outp[j * 2 + 0] = c_cos[j];
      outp[j * 2 + 1] = c_sin[j];
    }
  }
  if (lane == 16) {
    #pragma unroll
    for (int j = 0; j < 8; ++j) {
      outp[(8 + j) * 2 + 0] = c_cos[j];
      outp[(8 + j) * 2 + 1] = c_sin[j];
    }
  }
}

// Kernel 2: combine T-slices -> plv -> sqrt(plv+eps) (triad diag == 1),
// mean over F, clamp, sigmoid-hysteresis EMA update per batch.
__global__ __launch_bounds__(256)
void k2_finalize(const float* __restrict__ part,
                 const float* __restrict__ prev_coh,
                 const float* __restrict__ prev_alpha,
                 float* __restrict__ out) {
  __shared__ float vals[B_ * F_];
  const int tid = threadIdx.x;

  for (int p = tid; p < B_ * F_; p += 256) {
    const int b   = p >> 6;
    const int f   = p & 63;
    const int fg  = f >> 4;
    const int row = f & 15;
    float sc = 0.0f, ss = 0.0f;
    #pragma unroll
    for (int ts = 0; ts < TS_; ++ts) {
      const float* q =
          part + ((size_t)((b * FG_ + fg) * TS_ + ts) * 16 + row) * 2;
      sc += q[0];
      ss += q[1];
    }
    const float mc  = sc * (1.0f / (float)T_);
    const float ms  = ss * (1.0f / (float)T_);
    const float plv = sqrtf(mc * mc + ms * ms);
    vals[p] = sqrtf(plv + EPS_);
  }
  __syncthreads();

  if (tid < B_) {
    float acc = 0.0f;
    #pragma unroll
    for (int f = 0; f < F_; ++f) acc += vals[tid * F_ + f];
    const float coh = fminf(fmaxf(acc * (1.0f / (float)F_), 0.0f), 1.0f);
    const float pc  = prev_coh[tid];
    const float pa  = prev_alpha[tid];
    const float vel = coh - pc;
    const float sg  = 1.0f / (1.0f + expf(-(SIG_SLOPE_ * fabsf(vel) - SIG_OFF_)));
    const float target = ALPHA_MIN_ + (ALPHA_MAX_ - ALPHA_MIN_) * sg;
    const float alpha  = pa + BETA_ * (target - pa);
    out[tid] = alpha * coh + (1.0f - alpha) * pc;
  }
}

extern "C" void kernel_launch(void* const* d_in, const int* in_sizes, int n_in,
                              void* d_out, int out_size, void* d_ws, size_t ws_size,
                              hipStream_t stream) {
  const float* phases     = (const float*)d_in[0];
  const float* prev_coh   = (const float*)d_in[1];
  const float* prev_alpha = (const float*)d_in[2];
  float* out  = (float*)d_out;
  float* part = (float*)d_ws;  // needs B*FG*TS*16*2 floats = 32 KiB

  // 256 waves total = 8 waves/block * 32 blocks
  k1_plv_partial<<<32, 256, 0, stream>>>(phases, part);
  k2_finalize<<<1, 256, 0, stream>>>(part, prev_coh, prev_alpha, out);
}